// MatchingLayer_41953240547970
// MI455X (gfx1250) — compile-verified
//
#include <hip/hip_runtime.h>
#include <math.h>

#define EPS 1e-8f

// ---- CDNA5 WMMA fp32 16x16x4 ----------------------------------------------
typedef __attribute__((ext_vector_type(2))) float v2f;
typedef __attribute__((ext_vector_type(8))) float v8f;

__device__ __forceinline__ v8f wmma4(v2f a, v2f b, v8f c) {
  // (neg_a, A, neg_b, B, c_mod, C, reuse_a, reuse_b)
  return __builtin_amdgcn_wmma_f32_16x16x4_f32(false, a, false, b, (short)0, c,
                                               false, false);
}
// Fragment layout (wave32):
//  A 16x4:  lane = M (0..15 per half), half selects K base {0,2}; v0=K, v1=K+1
//  B 4x16:  lane = N, half selects K base {0,2}; v0=K row, v1=K+1 row
//  C 16x16: lane 0-15 N=lane M=v, lane 16-31 N=lane-16 M=v+8

// ---------------------------------------------------------------------------
// K1: pack q1/q2 into K-pair transposed layout:
//     qP[d][b][hp][i][c] = q[b][i][d*256 + 2*hp + c]   dims [2][16][128][128][2]
__global__ void pack_kernel(const float* __restrict__ q1,
                            const float* __restrict__ q2,
                            float* __restrict__ q1P, float* __restrict__ q2P) {
  __shared__ float tile[32][33];
  int blk = blockIdx.x;
  int ht = blk & 7;  blk >>= 3;   // h-tile (32 wide)
  int it = blk & 3;  blk >>= 2;   // i-tile (32 wide)
  int b  = blk & 15; blk >>= 4;
  int d  = blk & 1;  blk >>= 1;
  int src = blk;
  const float* q = src ? q2 : q1;
  float* out = src ? q2P : q1P;
  int t = threadIdx.x;
  int r  = t >> 3;          // 0..31 local i
  int c4 = (t & 7) * 4;     // 0..28 local h
  int i0 = it * 32, h0 = ht * 32;
  float4 v = *(const float4*)(q + ((size_t)(b * 128 + i0 + r)) * 512 + d * 256 + h0 + c4);
  tile[r][c4 + 0] = v.x; tile[r][c4 + 1] = v.y;
  tile[r][c4 + 2] = v.z; tile[r][c4 + 3] = v.w;
  __syncthreads();
  int hp_l = t >> 4;        // 0..15 local h-pair
  int i_l  = t & 15;
  int dp = d * 16 + b;
  #pragma unroll
  for (int rep = 0; rep < 2; ++rep) {
    int i  = i_l + rep * 16;
    int hp = (h0 >> 1) + hp_l;
    float2 w2 = make_float2(tile[i][2 * hp_l], tile[i][2 * hp_l + 1]);
    *(float2*)(out + ((size_t)(dp * 128 + hp)) * 256 + (size_t)(i0 + i) * 2) = w2;
  }
}

// ---------------------------------------------------------------------------
// K2: plain norms n1[d][b][i], n2[d][b][j]
__global__ void norm_kernel(const float* __restrict__ q1,
                            const float* __restrict__ q2,
                            float* __restrict__ n1, float* __restrict__ n2) {
  int t = blockIdx.x * blockDim.x + threadIdx.x;   // 8192
  int row = t & 127; int b = (t >> 7) & 15; int d = (t >> 11) & 1; int src = t >> 12;
  const float* p = (src ? q2 : q1) + ((size_t)(b * 128 + row)) * 512 + d * 256;
  float acc = 0.f;
  for (int h = 0; h < 256; h += 4) {
    float4 v = *(const float4*)(p + h);
    acc += v.x * v.x + v.y * v.y + v.z * v.z + v.w * v.w;
  }
  (src ? n2 : n1)[(d * 16 + b) * 128 + row] = sqrtf(acc);
}

// ---------------------------------------------------------------------------
// K3: weighted norms with W[2+d]:  wn[src][d][b][row][p]
__global__ void wnorm_kernel(const float* __restrict__ q1,
                             const float* __restrict__ q2,
                             const float* __restrict__ W,
                             float* __restrict__ wn_a, float* __restrict__ wn_c) {
  int t = blockIdx.x * blockDim.x + threadIdx.x;   // 131072
  int p = t & 15; int row = (t >> 4) & 127; int b = (t >> 11) & 15;
  int d = (t >> 15) & 1; int src = t >> 16;
  const float* x = (src ? q2 : q1) + ((size_t)(b * 128 + row)) * 512 + d * 256;
  const float* w = W + ((size_t)((2 + d) * 16 + p)) * 256;
  float acc = 0.f;
  for (int h = 0; h < 256; h += 4) {
    float4 xv = *(const float4*)(x + h);
    float4 wv = *(const float4*)(w + h);
    acc += xv.x * xv.x * wv.x * wv.x + xv.y * xv.y * wv.y * wv.y
         + xv.z * xv.z * wv.z * wv.z + xv.w * xv.w * wv.w * wv.w;
  }
  (src ? wn_c : wn_a)[((size_t)((d * 16 + b) * 128 + row)) * 16 + p] = sqrtf(acc);
}

// ---------------------------------------------------------------------------
// K4: attention GEMM (WMMA): att = (q1 . q2^T) / div_small(n1*n2)
//     stored K-pair packed for reuse as WMMA-A: attP[d][b][j/2][i][j&1]
__global__ void att_kernel(const float* __restrict__ q1P,
                           const float* __restrict__ q2P,
                           const float* __restrict__ n1,
                           const float* __restrict__ n2,
                           float* __restrict__ attP) {
  int blk = blockIdx.x;                 // 256 = 8 it * 16 b * 2 d
  int it = blk & 7; blk >>= 3;
  int b  = blk & 15; blk >>= 4;
  int d  = blk;
  int lane = threadIdx.x & 31;
  int wave = threadIdx.x >> 5;          // j-tile
  int dp = d * 16 + b;
  int i0 = it * 16, j0 = wave * 16;
  int m = lane & 15, hi = lane >> 4;
  const float* Abase = q1P + (size_t)dp * 32768 + (size_t)(i0 + m) * 2;
  const float* Bbase = q2P + (size_t)dp * 32768 + (size_t)(j0 + m) * 2;
  v8f c = {};
  for (int kk = 0; kk < 256; kk += 4) {
    int hp = (kk >> 1) + hi;
    v2f a  = *(const v2f*)(Abase + (size_t)hp * 256);
    v2f bb = *(const v2f*)(Bbase + (size_t)hp * 256);
    c = wmma4(a, bb, c);
  }
  int mb = hi * 8;
  const float* n1p = n1 + dp * 128;
  float nj = n2[dp * 128 + j0 + m];
  float* ad = attP + (size_t)dp * 16384;
  int j = j0 + m;
  size_t jbase = (size_t)(j >> 1) * 256 + (size_t)(j & 1);
  #pragma unroll
  for (int v = 0; v < 8; ++v) {
    int i = i0 + mb + v;
    float den = n1p[i] * nj;
    float ds = den > EPS ? den : EPS;   // div_small
    ad[jbase + (size_t)i * 2] = c[v] / ds;
  }
}

// ---------------------------------------------------------------------------
// K5: T[d][b][i] = sum_j att
__global__ void rowsum_kernel(const float* __restrict__ attP, float* __restrict__ T) {
  int t = blockIdx.x * blockDim.x + threadIdx.x;   // 4096
  int i = t & 127; int b = (t >> 7) & 15; int d = t >> 11;
  int dp = d * 16 + b;
  const float* ap = attP + (size_t)dp * 16384 + (size_t)i * 2;
  float acc = 0.f;
  for (int jp = 0; jp < 64; ++jp) {
    float2 v = *(const float2*)(ap + (size_t)jp * 256);
    acc += v.x + v.y;
  }
  T[dp * 128 + i] = acc;
}

// ---------------------------------------------------------------------------
// K6: cos_maxpool (big WMMA): per (d,b,p,itile) block, wave = jtile,
//     A = q1*W[2+d,p]^2 folded into fragment; block-wide max over j.
__global__ void maxpool_kernel(const float* __restrict__ q1P,
                               const float* __restrict__ q2P,
                               const float* __restrict__ W,
                               const float* __restrict__ wn_a,
                               const float* __restrict__ wn_c,
                               float* __restrict__ out) {
  __shared__ float red[8][16];
  int blk = blockIdx.x;                 // 4096 = 8 it * 16 p * 16 b * 2 d
  int it = blk & 7;  blk >>= 3;
  int p  = blk & 15; blk >>= 4;
  int b  = blk & 15; blk >>= 4;
  int d  = blk;
  int lane = threadIdx.x & 31;
  int wave = threadIdx.x >> 5;          // j-tile
  int dp = d * 16 + b;
  int i0 = it * 16, j0 = wave * 16;
  int m = lane & 15, hi = lane >> 4;
  const float* Abase = q1P + (size_t)dp * 32768 + (size_t)(i0 + m) * 2;
  const float* Bbase = q2P + (size_t)dp * 32768 + (size_t)(j0 + m) * 2;
  const float* wp = W + ((size_t)((2 + d) * 16 + p)) * 256;
  v8f c = {};
  for (int kk = 0; kk < 256; kk += 4) {
    int hp = (kk >> 1) + hi;
    float2 w = *(const float2*)(wp + (size_t)hp * 2);   // W at h, h+1 (uniform/half)
    v2f a = *(const v2f*)(Abase + (size_t)hp * 256);
    a.x *= w.x * w.x;
    a.y *= w.y * w.y;
    v2f bb = *(const v2f*)(Bbase + (size_t)hp * 256);
    c = wmma4(a, bb, c);
  }
  int mb = hi * 8;
  float ncj = wn_c[((size_t)(dp * 128 + j0 + m)) * 16 + p];
  float mx[8];
  #pragma unroll
  for (int v = 0; v < 8; ++v) {
    int i = i0 + mb + v;
    float den = wn_a[((size_t)(dp * 128 + i)) * 16 + p] * ncj;
    float r = c[v] / fmaxf(den, EPS);
    for (int s = 1; s < 16; s <<= 1) r = fmaxf(r, __shfl_xor(r, s, 16));  // max over N
    mx[v] = r;
  }
  if ((lane & 15) == 0) {
    #pragma unroll
    for (int v = 0; v < 8; ++v) red[wave][mb + v] = mx[v];
  }
  __syncthreads();
  if (threadIdx.x < 16) {
    int mm = threadIdx.x;
    float r = red[0][mm];
    #pragma unroll
    for (int w = 1; w < 8; ++w) r = fmaxf(r, red[w][mm]);
    out[((size_t)(b * 128 + i0 + mm)) * 128 + 32 + 16 * d + p] = r;
  }
}

// ---------------------------------------------------------------------------
// K7: mean = (att @ q2_d) / div_small(T_fw)  (reference uses T_fw for BOTH dirs)
__global__ void mean_kernel(const float* __restrict__ attP,
                            const float* __restrict__ q2,
                            const float* __restrict__ T,
                            float* __restrict__ mean) {
  int blk = blockIdx.x;                 // 256 = 8 it * 16 b * 2 d
  int it = blk & 7; blk >>= 3;
  int b  = blk & 15; blk >>= 4;
  int d  = blk;
  int lane = threadIdx.x & 31;
  int wave = threadIdx.x >> 5;
  int dp = d * 16 + b;
  int i0 = it * 16;
  int m = lane & 15, hi = lane >> 4, mb = hi * 8;
  const float* Abase = attP + (size_t)dp * 16384 + (size_t)(i0 + m) * 2;
  const float* Tp = T + b * 128;        // forward-direction sums (reference quirk)
  for (int nt = wave; nt < 16; nt += 8) {
    int h0 = nt * 16;
    const float* Bbase = q2 + (size_t)b * 65536 + d * 256 + h0 + m;
    v8f c = {};
    for (int kk = 0; kk < 128; kk += 4) {
      int jp = (kk >> 1) + hi;
      int kb = kk + 2 * hi;
      v2f a = *(const v2f*)(Abase + (size_t)jp * 256);
      v2f bb;
      bb.x = Bbase[(size_t)kb * 512];
      bb.y = Bbase[(size_t)(kb + 1) * 512];
      c = wmma4(a, bb, c);
    }
    #pragma unroll
    for (int v = 0; v < 8; ++v) {
      int i = i0 + mb + v;
      float t = Tp[i];
      float ts = t > EPS ? t : EPS;     // div_small
      mean[((size_t)(dp * 128 + i)) * 256 + h0 + m] = c[v] / ts;
    }
  }
}

// ---------------------------------------------------------------------------
// K8: max_at[d][b][i][h] = max_j att[i][j]*q2_d[j][h]
__global__ void maxat_kernel(const float* __restrict__ attP,
                             const float* __restrict__ q2,
                             float* __restrict__ maxat) {
  int t = blockIdx.x * blockDim.x + threadIdx.x;   // 1048576
  int h = t & 255; int i = (t >> 8) & 127; int b = (t >> 15) & 15; int d = t >> 19;
  int dp = d * 16 + b;
  const float* ap = attP + (size_t)dp * 16384 + (size_t)i * 2;
  const float* qp = q2 + (size_t)b * 65536 + d * 256 + h;
  float mx = -INFINITY;
  for (int jp = 0; jp < 64; ++jp) {
    float2 av = *(const float2*)(ap + (size_t)jp * 256);
    float q0 = qp[(size_t)(2 * jp) * 512];
    float q1v = qp[(size_t)(2 * jp + 1) * 512];
    mx = fmaxf(mx, av.x * q0);
    mx = fmaxf(mx, av.y * q1v);
  }
  maxat[((size_t)(dp * 128 + i)) * 256 + h] = mx;
}

// ---------------------------------------------------------------------------
// K9: per-(b,i,p) weighted cosines: full_fw/bw, attn_fw/bw, maxattn_fw/bw
__global__ void cosfull_kernel(const float* __restrict__ q1,
                               const float* __restrict__ q2,
                               const float* __restrict__ W,
                               const float* __restrict__ mean,
                               const float* __restrict__ maxat,
                               float* __restrict__ out) {
  int t = blockIdx.x * blockDim.x + threadIdx.x;   // 32768 per job
  int job = blockIdx.y;                             // 0..5
  int p = t & 15; int i = (t >> 4) & 127; int b = t >> 11;
  int d = job & 1;
  int k, outoff;
  const float* partner;
  switch (job) {
    case 0:  k = 0; outoff = 0;   partner = q2 + ((size_t)(b * 128 + 127)) * 512;        break;
    case 1:  k = 1; outoff = 16;  partner = q2 + ((size_t)(b * 128 + 0)) * 512 + 256;    break;
    case 2:  k = 4; outoff = 64;  partner = mean  + ((size_t)(b * 128 + i)) * 256;       break;
    case 3:  k = 5; outoff = 80;  partner = mean  + ((size_t)((16 + b) * 128 + i)) * 256; break;
    case 4:  k = 6; outoff = 96;  partner = maxat + ((size_t)(b * 128 + i)) * 256;       break;
    default: k = 7; outoff = 112; partner = maxat + ((size_t)((16 + b) * 128 + i)) * 256; break;
  }
  const float* x = q1 + ((size_t)(b * 128 + i)) * 512 + d * 256;
  const float* w = W + (size_t)k * 4096 + p * 256;
  float num = 0.f, na = 0.f, nc = 0.f;
  for (int h = 0; h < 256; h += 4) {
    float4 xv = *(const float4*)(x + h);
    float4 pv = *(const float4*)(partner + h);
    float4 wv = *(const float4*)(w + h);
    float w2x = wv.x * wv.x, w2y = wv.y * wv.y, w2z = wv.z * wv.z, w2w = wv.w * wv.w;
    num += xv.x * pv.x * w2x + xv.y * pv.y * w2y + xv.z * pv.z * w2z + xv.w * pv.w * w2w;
    na  += xv.x * xv.x * w2x + xv.y * xv.y * w2y + xv.z * xv.z * w2z + xv.w * xv.w * w2w;
    nc  += pv.x * pv.x * w2x + pv.y * pv.y * w2y + pv.z * pv.z * w2z + pv.w * pv.w * w2w;
  }
  float den = sqrtf(na) * sqrtf(nc);
  out[((size_t)(b * 128 + i)) * 128 + outoff + p] = num / fmaxf(den, EPS);
}

// ---------------------------------------------------------------------------
extern "C" void kernel_launch(void* const* d_in, const int* in_sizes, int n_in,
                              void* d_out, int out_size, void* d_ws, size_t ws_size,
                              hipStream_t stream) {
  const float* q1 = (const float*)d_in[0];   // (16,128,512)
  const float* q2 = (const float*)d_in[1];   // (16,128,512)
  const float* W  = (const float*)d_in[2];   // (8,16,256)
  float* out = (float*)d_out;                // (16,128,128)

  float* ws    = (float*)d_ws;
  float* q1P   = ws;                 // [2][16][128][128][2] = 1048576
  float* q2P   = q1P + 1048576;      // 1048576
  float* n1    = q2P + 1048576;      // 4096
  float* n2    = n1 + 4096;          // 4096
  float* wn_a  = n2 + 4096;          // 65536
  float* wn_c  = wn_a + 65536;       // 65536
  float* attP  = wn_c + 65536;       // [2][16][64][128][2] = 524288
  float* T     = attP + 524288;      // 4096
  float* mean  = T + 4096;           // 1048576
  float* maxat = mean + 1048576;     // 1048576   (total ~19.5 MB)

  pack_kernel   <<<2048, 256, 0, stream>>>(q1, q2, q1P, q2P);
  norm_kernel   <<<32,   256, 0, stream>>>(q1, q2, n1, n2);
  wnorm_kernel  <<<512,  256, 0, stream>>>(q1, q2, W, wn_a, wn_c);
  att_kernel    <<<256,  256, 0, stream>>>(q1P, q2P, n1, n2, attP);
  rowsum_kernel <<<16,   256, 0, stream>>>(attP, T);
  maxpool_kernel<<<4096, 256, 0, stream>>>(q1P, q2P, W, wn_a, wn_c, out);
  mean_kernel   <<<256,  256, 0, stream>>>(attP, q2, T, mean);
  maxat_kernel  <<<4096, 256, 0, stream>>>(attP, q2, maxat);
  cosfull_kernel<<<dim3(128, 6), 256, 0, stream>>>(q1, q2, W, mean, maxat, out);
}